// ResiduesNetwork_27058293965309
// MI455X (gfx1250) — compile-verified
//
#include <hip/hip_runtime.h>
#include <hip/hip_bf16.h>

typedef __attribute__((ext_vector_type(2))) float v2f;
typedef __attribute__((ext_vector_type(8))) float v8f;

#define N_RES 800   // N1 == N2 == 800
#define KNBR  10

// ---------------------------------------------------------------------------
// f32 WMMA helper: D(16x16,f32) = A(16x4,f32) * B(4x16,f32) + C
// A frag: lanes 0-15 hold M=lane with K=k+0/k+1 in .x/.y; lanes 16-31 K=k+2/k+3
// B frag: mirror (K rows striped), C/D: VGPR r holds M=r (lanes0-15) / M=r+8
// ---------------------------------------------------------------------------
__device__ __forceinline__ v8f wmma_f32x4(v2f a, v2f b, v8f c) {
  return __builtin_amdgcn_wmma_f32_16x16x4_f32(
      /*neg_a=*/false, a, /*neg_b=*/false, b,
      /*c_mod=*/(short)0, c, /*reuse_a=*/false, /*reuse_b=*/false);
}

__device__ __forceinline__ float frelu(float x) { return x > 0.f ? x : 0.f; }

// ---------------------------------------------------------------------------
// Generic fp32 GEMM: C[M,N] = A[M,K] @ B[K,N]. One 16x16 tile per wave,
// 8 waves per block. M%16==0, N%16==0, K%4==0 (true for all our shapes).
// ---------------------------------------------------------------------------
__global__ __launch_bounds__(256) void gemm_wmma_f32(
    const float* __restrict__ A, const float* __restrict__ B,
    float* __restrict__ C, int M, int K, int N, int lda, int ldb, int ldc) {
  const int lane = threadIdx.x & 31;
  const int wave = threadIdx.x >> 5;
  const int ntiles = N >> 4;
  const int mtiles = M >> 4;
  const int t = blockIdx.x * 8 + wave;
  if (t >= mtiles * ntiles) return;           // uniform per wave -> EXEC stays full
  const int mt = t / ntiles;
  const int nt = t - mt * ntiles;
  const int l15  = lane & 15;
  const int kOff = (lane >> 4) << 1;          // 0 or 2
  const int row  = mt * 16 + l15;
  const int col  = nt * 16 + l15;
  const float* Arow = A + row * lda;

  v8f acc = {};
  for (int k = 0; k < K; k += 4) {
    // contiguous pair -> single global_load_b64
    v2f a = *(const v2f*)(Arow + k + kOff);
    v2f b;
    b.x = B[(k + kOff) * ldb + col];
    b.y = B[(k + kOff + 1) * ldb + col];
    acc = wmma_f32x4(a, b, acc);
  }

  const int rbase = mt * 16 + ((lane >> 4) << 3);
#pragma unroll
  for (int r = 0; r < 8; ++r)
    C[(rbase + r) * ldc + col] = acc[r];
}

// ---------------------------------------------------------------------------
// GNN neighbor aggregation: out = relu(res + sum_k base[nb[n,k]] / max(cnt,1))
// One block per residue, one thread per feature.
// ---------------------------------------------------------------------------
__global__ void gnn_aggregate(const float* __restrict__ res,
                              const float* __restrict__ base,
                              const int* __restrict__ nb,
                              float* __restrict__ out, int F) {
  const int n = blockIdx.x;
  const int f = threadIdx.x;
  const int* nbr = nb + n * KNBR;
  float s = 0.f;
  int cnt = 0;
#pragma unroll
  for (int k = 0; k < KNBR; ++k) {
    const int idx = nbr[k];
    if (idx >= 0) { ++cnt; s += base[idx * F + f]; }
  }
  const float norm = (float)(cnt > 0 ? cnt : 1);
  const float v = res[n * F + f] + s / norm;
  out[n * F + f] = frelu(v);
}

// ---------------------------------------------------------------------------
// One-off transpose: fW1 [256,64] -> fW1t [64,256] so WMMA B-fragments become
// contiguous 8-byte loads in the pair kernel's inner loop.
// ---------------------------------------------------------------------------
__global__ void transpose_w1(const float* __restrict__ fW1,
                             float* __restrict__ fW1t) {
  const int idx = blockIdx.x * 256 + threadIdx.x;   // 0 .. 16383
  const int k = idx >> 6, n = idx & 63;
  fW1t[n * 256 + k] = fW1[idx];
}

// ---------------------------------------------------------------------------
// Fused all-pairs MLP. Block = 16 i's x 16 j's = 256 pairs (M=256).
//   h0[q,:] = relu(P1[i] + P2[j] + fb0)   (built on the fly as WMMA A frags)
//   h1 = relu(h0 @ fW1 + fb1)             (f32 WMMA, K=256, N=64)
//   out[i*800+j] = h1 . fW2 + fb2         (half-wave shuffle reduction)
// Wave w owns M-tiles {2w, 2w+1} x all 4 N-tiles -> 8 accumulators.
// Pair q = ii*16 + jj, so for tile mt: ii = mt (wave-uniform), jj = lane&15.
// ---------------------------------------------------------------------------
__global__ __launch_bounds__(256) void pair_mlp(
    const float* __restrict__ P1, const float* __restrict__ P2,
    const float* __restrict__ fb0, const float* __restrict__ fW1t,
    const float* __restrict__ fb1, const float* __restrict__ fW2,
    const float* __restrict__ fb2, float* __restrict__ out) {
  __shared__ float sP1[16 * 256];
  __shared__ float sP2[16 * 256];
  __shared__ float sB0[256];
  __shared__ float sB1[64];
  __shared__ float sW2[64];

  const int tid  = threadIdx.x;
  const int lane = tid & 31;
  const int wave = tid >> 5;
  const int i0 = blockIdx.x * 16;
  const int j0 = blockIdx.y * 16;

  // Warm L1/L2 with fW1t (64KB total, one 256B-stride line per thread)
  __builtin_prefetch(fW1t + (tid << 6), 0, 3);

  // 16x256 tiles are contiguous (ld == 256): b128 staging copies
  {
    const float4* gp1 = (const float4*)(P1 + i0 * 256);
    const float4* gp2 = (const float4*)(P2 + j0 * 256);
    float4* lp1 = (float4*)sP1;
    float4* lp2 = (float4*)sP2;
#pragma unroll
    for (int idx = tid; idx < 1024; idx += 256) {
      lp1[idx] = gp1[idx];
      lp2[idx] = gp2[idx];
    }
  }
  sB0[tid] = fb0[tid];
  if (tid < 64) { sB1[tid] = fb1[tid]; sW2[tid] = fW2[tid]; }
  __syncthreads();

  const int l15  = lane & 15;
  const int half = lane >> 4;
  const int kOff = half << 1;           // 0 or 2 (always even -> 8B aligned)
  const int m0   = wave * 2;            // first of two M-tiles for this wave

  v8f acc[2][4];
#pragma unroll
  for (int m = 0; m < 2; ++m)
#pragma unroll
    for (int nt = 0; nt < 4; ++nt) acc[m][nt] = (v8f){};

  const v2f* p2v  = (const v2f*)(sP2 + l15 * 256);
  const v2f* p1a  = (const v2f*)(sP1 + m0 * 256);        // wave-uniform rows
  const v2f* p1b  = (const v2f*)(sP1 + (m0 + 1) * 256);
  const v2f* b0v  = (const v2f*)sB0;
  const v2f* w1t  = (const v2f*)fW1t;                    // [n][k] pairs
  const int  nrow = l15 * 128;                           // v2f stride per n-col

  for (int k = 0; k < 256; k += 4) {
    const int kh = (k + kOff) >> 1;                      // v2f index
    const v2f bias = b0v[kh];
    const v2f p2   = p2v[kh];
    const v2f pa   = p1a[kh];
    const v2f pb   = p1b[kh];
    v2f a0, a1;
    a0.x = frelu(pa.x + p2.x + bias.x);
    a0.y = frelu(pa.y + p2.y + bias.y);
    a1.x = frelu(pb.x + p2.x + bias.x);
    a1.y = frelu(pb.y + p2.y + bias.y);
#pragma unroll
    for (int nt = 0; nt < 4; ++nt) {
      const v2f b = w1t[(nt * 16) * 128 + nrow + kh];    // one global_load_b64
      acc[0][nt] = wmma_f32x4(a0, b, acc[0][nt]);
      acc[1][nt] = wmma_f32x4(a1, b, acc[1][nt]);
    }
  }

  // Epilogue: relu(+fb1), dot with fW2 (reduce 16 lanes per half), +fb2.
  float w2v[4], b1v[4];
#pragma unroll
  for (int nt = 0; nt < 4; ++nt) {
    w2v[nt] = sW2[nt * 16 + l15];
    b1v[nt] = sB1[nt * 16 + l15];
  }
  const float fb2v = fb2[0];

#pragma unroll
  for (int m = 0; m < 2; ++m) {
    const int mt = m0 + m;
#pragma unroll
    for (int r = 0; r < 8; ++r) {
      float s = 0.f;
#pragma unroll
      for (int nt = 0; nt < 4; ++nt)
        s += frelu(acc[m][nt][r] + b1v[nt]) * w2v[nt];
      // sum the 16 N-lanes of each half (xor of bits 0-3 stays within half)
      s += __shfl_xor(s, 1, 32);
      s += __shfl_xor(s, 2, 32);
      s += __shfl_xor(s, 4, 32);
      s += __shfl_xor(s, 8, 32);
      if (l15 == 0) {
        const int jj = r + (half << 3);     // q & 15
        out[(i0 + mt) * N_RES + j0 + jj] = s + fb2v;
      }
    }
  }
}

// ---------------------------------------------------------------------------
extern "C" void kernel_launch(void* const* d_in, const int* in_sizes, int n_in,
                              void* d_out, int out_size, void* d_ws, size_t ws_size,
                              hipStream_t stream) {
  const float* Z1   = (const float*)d_in[0];
  const int*   nb1  = (const int*)  d_in[1];
  const float* Z2   = (const float*)d_in[2];
  const int*   nb2  = (const int*)  d_in[3];
  const float* Wr0  = (const float*)d_in[4];
  const float* Wnr0 = (const float*)d_in[5];
  const float* Wr1  = (const float*)d_in[6];
  const float* Wnr1 = (const float*)d_in[7];
  const float* fW0  = (const float*)d_in[8];
  const float* fb0  = (const float*)d_in[9];
  const float* fW1  = (const float*)d_in[10];
  const float* fb1  = (const float*)d_in[11];
  const float* fW2  = (const float*)d_in[12];
  const float* fb2  = (const float*)d_in[13];
  float* out = (float*)d_out;

  float* ws   = (float*)d_ws;
  float* res  = ws;                    // 800*256
  float* base = res  + N_RES * 256;    // 800*256
  float* h0   = base + N_RES * 256;    // 800*256 (layer-0 output, reused)
  float* x1   = h0   + N_RES * 256;    // 800*128
  float* x2   = x1   + N_RES * 128;    // 800*128
  float* P1   = x2   + N_RES * 128;    // 800*256
  float* P2   = P1   + N_RES * 256;    // 800*256
  float* fW1t = P2   + N_RES * 256;    // 64*256

  auto run_protein = [&](const float* Z, const int* nb, float* xout) {
    // layer 0: 800x1024 @ 1024x256 -> 50*16 tiles / 8 waves = 100 blocks
    gemm_wmma_f32<<<100, 256, 0, stream>>>(Z, Wr0,  res,  800, 1024, 256, 1024, 256, 256);
    gemm_wmma_f32<<<100, 256, 0, stream>>>(Z, Wnr0, base, 800, 1024, 256, 1024, 256, 256);
    gnn_aggregate<<<N_RES, 256, 0, stream>>>(res, base, nb, h0, 256);
    // layer 1: 800x256 @ 256x128 -> 50*8 tiles / 8 = 50 blocks
    gemm_wmma_f32<<<50, 256, 0, stream>>>(h0, Wr1,  res,  800, 256, 128, 256, 128, 128);
    gemm_wmma_f32<<<50, 256, 0, stream>>>(h0, Wnr1, base, 800, 256, 128, 256, 128, 128);
    gnn_aggregate<<<N_RES, 128, 0, stream>>>(res, base, nb, xout, 128);
  };
  run_protein(Z1, nb1, x1);
  run_protein(Z2, nb2, x2);

  // Decompose first pair layer: pairs@fW0 = x1@fW0[:128] + x2@fW0[128:]
  gemm_wmma_f32<<<100, 256, 0, stream>>>(x1, fW0,             P1, 800, 128, 256, 128, 256, 256);
  gemm_wmma_f32<<<100, 256, 0, stream>>>(x2, fW0 + 128 * 256, P2, 800, 128, 256, 128, 256, 256);

  // Transpose fW1 for contiguous B-fragment loads
  transpose_w1<<<64, 256, 0, stream>>>(fW1, fW1t);

  // Fused pair MLP over all 800x800 pairs (50x50 blocks of 16x16 pairs)
  pair_mlp<<<dim3(50, 50), 256, 0, stream>>>(P1, P2, fb0, fW1t, fb1, fW2, fb2, out);
}